// ConvexUpsampler_57131654971446
// MI455X (gfx1250) — compile-verified
//
#include <hip/hip_runtime.h>

// MI455X / gfx1250, wave32. fp32 WMMA (V_WMMA_F32_16X16X4_F32) implicit-GEMM
// convex upsampler. Compute-bound (~400 FLOP/byte), fp32 matrix pipe chosen to
// match the fp32 reference exactly while still using the CDNA5 matrix units.

typedef __attribute__((ext_vector_type(2))) float v2f;
typedef __attribute__((ext_vector_type(8))) float v8f;

#define HID   128
#define HH    48
#define WW    48
#define BB    8
#define NMASK 576          // 9 * 8 * 8
#define EPSBN 1e-5f

// Workspace layout (floats):
//   Wt1 [9][128][128]   conv1 weights, tap-major, cout innermost (coalesced B loads)
//   Wt2 [128][576]      conv2 weights transposed (k-major)
//   Xn  [B][H][W][128]  stage-1 activation, NHWC (contiguous per 16-pixel tile)
#define WT1_ELEMS (9 * HID * HID)
#define WT2_ELEMS (HID * NMASK)

// ---------------------------------------------------------------------------
// Kernel 0: weight permutations (tiny, bandwidth-trivial).
// ---------------------------------------------------------------------------
__global__ void prep_weights(const float* __restrict__ w1,   // [128][128][3][3]
                             const float* __restrict__ w2,   // [576][128]
                             float* __restrict__ Wt1,
                             float* __restrict__ Wt2) {
    int idx = blockIdx.x * blockDim.x + threadIdx.x;
    if (idx < WT1_ELEMS) {
        int o   = idx & 127;
        int c   = (idx >> 7) & 127;
        int tap = idx >> 14;
        Wt1[idx] = w1[(o * HID + c) * 9 + tap];
    } else {
        int j = idx - WT1_ELEMS;
        if (j < WT2_ELEMS) {
            int n = j % NMASK;
            int k = j / NMASK;
            Wt2[j] = w2[n * HID + k];
        }
    }
}

// ---------------------------------------------------------------------------
// Stage 1: conv3x3 (implicit GEMM, K = 128*9 = 1152) + BN + LeakyReLU(0.1).
// Block = 16 pixels (one row segment) x 128 out channels. 8 waves, each owning
// one 16x16 output tile; all waves share the LDS feat halo tile.
// ---------------------------------------------------------------------------
__global__ void __launch_bounds__(256)
conv1_kernel(const float* __restrict__ feat,   // [B][128][48][48]
             const float* __restrict__ Wt1,    // [9][128][128]
             const float* __restrict__ gamma,
             const float* __restrict__ beta,
             const float* __restrict__ mean,
             const float* __restrict__ var,
             float* __restrict__ Xn) {         // [B][48][48][128] NHWC
    __shared__ float lfeat[HID * 3 * 18];      // [c][di][j], j = w0-1 .. w0+16

    const int bid = blockIdx.x;
    const int wt  = bid % 3;
    const int h   = (bid / 3) % HH;
    const int b   = bid / (3 * HH);
    const int w0  = wt * 16;
    const int tid = threadIdx.x;

    // Cooperative halo load (zero-padded), coalesced along j.
    for (int e = tid; e < HID * 54; e += 256) {
        int c   = e / 54;
        int rem = e % 54;
        int di  = rem / 18;
        int j   = rem % 18;
        int hh  = h + di - 1;
        int wp  = w0 - 1 + j;
        float v = 0.0f;
        if (hh >= 0 && hh < HH && wp >= 0 && wp < WW)
            v = feat[((b * HID + c) * HH + hh) * WW + wp];
        lfeat[e] = v;
    }
    __syncthreads();

    const int wave = tid >> 5;
    const int lane = tid & 31;
    const int half = lane >> 4;     // A/B: K = 2*half + r   (r = vector slot)
    const int lm   = lane & 15;     // A: M = lm, B: N = lm
    const int n0   = wave * 16;     // this wave's out-channel tile

    v8f acc = {0.f, 0.f, 0.f, 0.f, 0.f, 0.f, 0.f, 0.f};

    for (int tap = 0; tap < 9; ++tap) {
        const int di = tap / 3, dj = tap % 3;
        const float* __restrict__ lrow = &lfeat[di * 18 + dj + lm];   // + c*54
        const float* __restrict__ brow = &Wt1[tap * (HID * HID) + n0 + lm]; // + c*128
#pragma unroll 4
        for (int c0 = 0; c0 < HID; c0 += 4) {
            const int k0 = c0 + 2 * half;
            v2f a, bf;
            a.x  = lrow[(k0    ) * 54];
            a.y  = lrow[(k0 + 1) * 54];
            bf.x = brow[(k0    ) * HID];
            bf.y = brow[(k0 + 1) * HID];
            acc = __builtin_amdgcn_wmma_f32_16x16x4_f32(
                false, a, false, bf, (short)0, acc, false, false);
        }
    }

    // Fused BatchNorm + LeakyReLU, store NHWC (coalesced over lm = channel).
    const int ch = n0 + lm;
    const float sc = gamma[ch] * rsqrtf(var[ch] + EPSBN);
    const float sh = beta[ch] - mean[ch] * sc;
#pragma unroll
    for (int r = 0; r < 8; ++r) {
        const int m = r + 8 * half;   // C/D layout: M = r + 8*half
        float v = acc[r] * sc + sh;
        v = (v >= 0.0f) ? v : 0.1f * v;
        Xn[((b * HH + h) * WW + (w0 + m)) * HID + ch] = v;
    }
}

// ---------------------------------------------------------------------------
// Stage 2: conv1x1 128->576 (GEMM K=128) + bias + softmax over 9 + convex
// combination of the 3x3 flow neighborhood + pixel-shuffle store.
// 288 threads = 9 waves; wave w owns channels [w*64, w*64+64) == softmax slot
// k9 = w for all 64 (si,sj) positions -> softmax reduces across waves via LDS.
// ---------------------------------------------------------------------------
__global__ void __launch_bounds__(288)
conv2_upsample_kernel(const float* __restrict__ Xn,      // [B][48][48][128]
                      const float* __restrict__ Wt2,     // [128][576]
                      const float* __restrict__ bias,    // [576]
                      const float* __restrict__ flow_low,// [B][2][48][48]
                      float* __restrict__ out) {         // [B][2][384][384]
    __shared__ float lx[16 * HID];        //  8 KB activation tile
    __shared__ float lmask[16 * NMASK];   // 36 KB mask logits
    __shared__ float lflow[2 * 54];       // flow halo (already *8)

    const int bid = blockIdx.x;
    const int wt  = bid % 3;
    const int h   = (bid / 3) % HH;
    const int b   = bid / (3 * HH);
    const int w0  = wt * 16;
    const int tid = threadIdx.x;

    // 16 pixels x 128 ch are contiguous in NHWC -> fully coalesced bulk load.
    for (int e = tid; e < 16 * HID; e += 288)
        lx[e] = Xn[((b * HH + h) * WW + w0) * HID + e];

    for (int e = tid; e < 108; e += 288) {
        int xy  = e / 54;
        int rem = e % 54;
        int di  = rem / 18, j = rem % 18;
        int hh  = h + di - 1, wp = w0 - 1 + j;
        float v = 0.0f;
        if (hh >= 0 && hh < HH && wp >= 0 && wp < WW)
            v = 8.0f * flow_low[((b * 2 + xy) * HH + hh) * WW + wp];
        lflow[e] = v;
    }
    __syncthreads();

    const int wave = tid >> 5;      // 0..8  == softmax index k9
    const int lane = tid & 31;
    const int half = lane >> 4;
    const int lm   = lane & 15;
    const int nb   = wave * 64;     // base out-channel of this wave

    v8f acc[4];
#pragma unroll
    for (int t = 0; t < 4; ++t)
        acc[t] = (v8f){0.f, 0.f, 0.f, 0.f, 0.f, 0.f, 0.f, 0.f};

#pragma unroll 2
    for (int c0 = 0; c0 < HID; c0 += 4) {
        const int k0 = c0 + 2 * half;
        v2f a;
        a.x = lx[lm * HID + k0];
        a.y = lx[lm * HID + k0 + 1];
        const float* __restrict__ bp = &Wt2[k0 * NMASK + nb + lm];
#pragma unroll
        for (int t = 0; t < 4; ++t) {
            v2f bf;
            bf.x = bp[t * 16];
            bf.y = bp[NMASK + t * 16];
            acc[t] = __builtin_amdgcn_wmma_f32_16x16x4_f32(
                false, a, false, bf, (short)0, acc[t], false, false);
        }
    }

    // Bias + scatter logits to LDS: lmask[pixel][ch].
#pragma unroll
    for (int t = 0; t < 4; ++t) {
        const int ch = nb + t * 16 + lm;
        const float bv = bias[ch];
#pragma unroll
        for (int r = 0; r < 8; ++r) {
            const int m = r + 8 * half;
            lmask[m * NMASK + ch] = acc[t][r] + bv;
        }
    }
    __syncthreads();

    // Softmax over k9 (stride-64 channels) + convex flow combination.
    for (int p = tid; p < 16 * 64; p += 288) {
        const int m = p >> 6;        // pixel in tile
        const int s = p & 63;        // si*8 + sj
        float l[9];
        float mx = -3.0e38f;
#pragma unroll
        for (int k = 0; k < 9; ++k) {
            l[k] = lmask[m * NMASK + k * 64 + s];
            mx = fmaxf(mx, l[k]);
        }
        float sum = 0.0f;
#pragma unroll
        for (int k = 0; k < 9; ++k) {
            l[k] = __expf(l[k] - mx);
            sum += l[k];
        }
        const float inv = 1.0f / sum;
        float fx = 0.0f, fy = 0.0f;
#pragma unroll
        for (int k = 0; k < 9; ++k) {
            const float wk = l[k] * inv;
            const int di = k / 3, dj = k % 3;
            fx += wk * lflow[     di * 18 + dj + m];
            fy += wk * lflow[54 + di * 18 + dj + m];
        }
        const int si  = s >> 3, sj = s & 7;
        const int row = h * 8 + si;
        const int col = (w0 + m) * 8 + sj;
        out[((b * 2 + 0) * 384 + row) * 384 + col] = fx;
        out[((b * 2 + 1) * 384 + row) * 384 + col] = fy;
    }
}

// ---------------------------------------------------------------------------
extern "C" void kernel_launch(void* const* d_in, const int* in_sizes, int n_in,
                              void* d_out, int out_size, void* d_ws, size_t ws_size,
                              hipStream_t stream) {
    const float* feat    = (const float*)d_in[0];
    const float* flow    = (const float*)d_in[1];
    const float* conv1_w = (const float*)d_in[2];
    const float* gamma   = (const float*)d_in[3];
    const float* beta    = (const float*)d_in[4];
    const float* mean    = (const float*)d_in[5];
    const float* var     = (const float*)d_in[6];
    const float* conv2_w = (const float*)d_in[7];
    const float* conv2_b = (const float*)d_in[8];
    float* out = (float*)d_out;

    float* Wt1 = (float*)d_ws;
    float* Wt2 = Wt1 + WT1_ELEMS;
    float* Xn  = Wt2 + WT2_ELEMS;   // total ws use ~10.3 MB

    const int ntiles = BB * HH * 3;  // 1152 spatial tiles of 16 pixels

    prep_weights<<<(WT1_ELEMS + WT2_ELEMS + 255) / 256, 256, 0, stream>>>(
        conv1_w, conv2_w, Wt1, Wt2);
    conv1_kernel<<<ntiles, 256, 0, stream>>>(
        feat, Wt1, gamma, beta, mean, var, Xn);
    conv2_upsample_kernel<<<ntiles, 288, 0, stream>>>(
        Xn, Wt2, conv2_b, flow, out);
}